// SRT_GT_31533649887821
// MI455X (gfx1250) — compile-verified
//
#include <hip/hip_runtime.h>
#include <hip/hip_bf16.h>

// MI455X (gfx1250) fused GNN message-passing step.
//   edge kernel : gather x[src] -> (16x128)@(128x128)^T via V_WMMA_F32_16X16X4_F32
//                 -> LayerNorm (cross-lane shfl reduce) -> atomic scatter into agg
//   node kernel : upd = relu(x + c*agg + xi*local); out = upd @ out_w^T + out_b + upd
// Each wave processes TWO 16-row tiles per pass so every LDS B-fragment feeds
// two WMMAs, and A/B fragments are double-buffered one K-step ahead so the
// compiler can batch s_wait_dscnt instead of stalling per WMMA.

#define D 128

typedef __attribute__((ext_vector_type(2))) float v2f;
typedef __attribute__((ext_vector_type(8))) float v8f;

__device__ __forceinline__ v8f wmma_f32(v2f a, v2f b, v8f c) {
    // D = A(16x4, f32) x B(4x16, f32) + C(16x16, f32)
    return __builtin_amdgcn_wmma_f32_16x16x4_f32(
        /*neg_a=*/false, a, /*neg_b=*/false, b,
        /*c_mod=*/(short)0, c, /*reuse_a=*/false, /*reuse_b=*/false);
}

// XOR-swizzled LDS address for a row-major 128x128 f32 matrix, 2-float granule.
// Keeps (k, k+1) contiguous (8B-aligned -> ds_load_b64) while spreading rows
// across banks so 16 lanes reading 16 rows at the same k are conflict-free.
__device__ __forceinline__ int sw_addr(int row, int col) {
    return (row << 7) + ((((col >> 1) ^ (row & 63)) << 1) | (col & 1));
}

__device__ __forceinline__ v2f ldsB(const float* sW, int kk, int half, int lm, int n) {
    const int row = (n << 4) + lm;              // output column N
    const int kB  = (kk << 2) + (half << 1);    // K offset for this lane half
    return *(const v2f*)&sW[sw_addr(row, kB)];
}

__global__ void zero_f4_kernel(float4* __restrict__ p, int n4) {
    for (int i = blockIdx.x * 256 + threadIdx.x; i < n4; i += gridDim.x * 256)
        p[i] = make_float4(0.f, 0.f, 0.f, 0.f);
}

// ---------------------------------------------------------------------------
// LayerNorm + gated scatter for one 16-edge tile held in WMMA C-layout.
// ---------------------------------------------------------------------------
__device__ __forceinline__ void ln_scatter(
    const v8f* __restrict__ acc, const int* __restrict__ dst_idx, int ebase,
    int half, int lm, const float* __restrict__ lg, const float* __restrict__ lb,
    float* __restrict__ agg)
{
    // Row M = j + 8*half lives in acc[*][j] across a 16-lane half: a shfl-xor
    // {1,2,4,8} tree yields per-row mean / rsigma directly in the right lanes.
    float mu[8], rs[8];
#pragma unroll
    for (int j = 0; j < 8; ++j) {
        float s = 0.f, q = 0.f;
#pragma unroll
        for (int n = 0; n < 8; ++n) {
            const float v = acc[n][j];
            s += v; q += v * v;
        }
#pragma unroll
        for (int m = 1; m < 16; m <<= 1) {
            s += __shfl_xor(s, m, 32);
            q += __shfl_xor(q, m, 32);
        }
        const float mean = s * (1.f / 128.f);
        mu[j] = mean;
        rs[j] = rsqrtf(fmaxf(q * (1.f / 128.f) - mean * mean, 0.f) + 1e-5f);
    }

    int drow[8];
#pragma unroll
    for (int j = 0; j < 8; ++j) drow[j] = dst_idx[ebase + j + (half << 3)];

#pragma unroll
    for (int n = 0; n < 8; ++n) {
        const int N = lm + (n << 4);
#pragma unroll
        for (int j = 0; j < 8; ++j) {
            const float v = (acc[n][j] - mu[j]) * rs[j] * lg[n] + lb[n];
            atomicAdd(&agg[(size_t)drow[j] * D + N], v);
        }
    }
}

// ---------------------------------------------------------------------------
// Edge kernel: two 16-edge tiles per wave per pass (B fragments reused 2x).
// ---------------------------------------------------------------------------
__global__ __launch_bounds__(256) void gnn_edge_kernel(
    const float* __restrict__ x,
    const int*   __restrict__ src_idx,
    const int*   __restrict__ dst_idx,
    const float* __restrict__ W_w,
    const float* __restrict__ W_b,
    const float* __restrict__ ln_g,
    const float* __restrict__ ln_b,
    float*       __restrict__ agg,
    int n_edges)
{
    __shared__ float sW[D * D];  // 64 KB, XOR-swizzled row-major W_w
    for (int i = threadIdx.x; i < D * D; i += 256) {
        int r = i >> 7, c = i & 127;
        sW[sw_addr(r, c)] = W_w[i];
    }
    __syncthreads();

    const int lane = threadIdx.x & 31;
    const int wave = threadIdx.x >> 5;
    const int half = lane >> 4;
    const int lm   = lane & 15;

    float wb[8], lg[8], lb[8];
#pragma unroll
    for (int n = 0; n < 8; ++n) {
        const int N = lm + (n << 4);
        wb[n] = W_b[N]; lg[n] = ln_g[N]; lb[n] = ln_b[N];
    }

    const int n_tiles = n_edges >> 4;
    const int n_pairs = (n_tiles + 1) >> 1;
    for (int pair = blockIdx.x * 8 + wave; pair < n_pairs; pair += gridDim.x * 8) {
        const int tile0  = pair << 1;
        const int tile1  = tile0 + 1;
        const bool has1  = tile1 < n_tiles;
        const int ebase0 = tile0 << 4;
        const int ebase1 = (has1 ? tile1 : tile0) << 4;  // clamp (waste, not fault)

        const int src0 = src_idx[ebase0 + lm];
        const int src1 = src_idx[ebase1 + lm];
        const float* arow0 = x + (size_t)src0 * D + (half << 1);
        const float* arow1 = x + (size_t)src1 * D + (half << 1);

        v8f acc0[8], acc1[8];
#pragma unroll
        for (int n = 0; n < 8; ++n) {
#pragma unroll
            for (int q = 0; q < 8; ++q) { acc0[n][q] = wb[n]; acc1[n][q] = wb[n]; }
        }

        // Software-pipelined GEMM: prefetch A/B fragments for kk+1 before the
        // 16 WMMAs of kk so DS/global waits overlap matrix-core work.
        v2f bb[2][8];
        v2f aa0[2], aa1[2];
#pragma unroll
        for (int n = 0; n < 8; ++n) bb[0][n] = ldsB(sW, 0, half, lm, n);
        aa0[0] = *(const v2f*)&arow0[0];
        aa1[0] = *(const v2f*)&arow1[0];

#pragma unroll 2
        for (int kk = 0; kk < 32; ++kk) {
            const int cur = kk & 1, nxt = cur ^ 1;
            if (kk < 31) {
#pragma unroll
                for (int n = 0; n < 8; ++n) bb[nxt][n] = ldsB(sW, kk + 1, half, lm, n);
                aa0[nxt] = *(const v2f*)&arow0[(kk + 1) << 2];
                aa1[nxt] = *(const v2f*)&arow1[(kk + 1) << 2];
            }
#pragma unroll
            for (int n = 0; n < 8; ++n) acc0[n] = wmma_f32(aa0[cur], bb[cur][n], acc0[n]);
#pragma unroll
            for (int n = 0; n < 8; ++n) acc1[n] = wmma_f32(aa1[cur], bb[cur][n], acc1[n]);
        }

        ln_scatter(acc0, dst_idx, ebase0, half, lm, lg, lb, agg);
        if (has1) ln_scatter(acc1, dst_idx, ebase1, half, lm, lg, lb, agg);
    }
}

// ---------------------------------------------------------------------------
// Node kernel: upd = relu(x + c_agg*agg + c_xi*local);  out = upd@W^T + b + upd
// ---------------------------------------------------------------------------
__device__ __forceinline__ float upd_elem(const float* __restrict__ x,
                                          const float* __restrict__ agg,
                                          const float* __restrict__ local,
                                          size_t idx, float ca, float cx) {
    return fmaxf(x[idx] + ca * agg[idx] + cx * local[idx], 0.f);
}

__device__ __forceinline__ v2f upd_frag(const float* __restrict__ x,
                                        const float* __restrict__ agg,
                                        const float* __restrict__ local,
                                        size_t o, float ca, float cx) {
    v2f a;
    a.x = upd_elem(x, agg, local, o + 0, ca, cx);
    a.y = upd_elem(x, agg, local, o + 1, ca, cx);
    return a;
}

__device__ __forceinline__ void node_store(
    const v8f* __restrict__ acc, const float* __restrict__ x,
    const float* __restrict__ agg, const float* __restrict__ local,
    float* __restrict__ out, int nbase, int half, int lm,
    const float* __restrict__ ob, float ca, float cx)
{
    const size_t gbase = (size_t)nbase * D;
#pragma unroll
    for (int n = 0; n < 8; ++n) {
        const int N = lm + (n << 4);
#pragma unroll
        for (int j = 0; j < 8; ++j) {
            const int M = j + (half << 3);
            const size_t idx = gbase + (size_t)M * D + N;
            const float u = upd_elem(x, agg, local, idx, ca, cx);
            out[idx] = acc[n][j] + u + ob[n];
        }
    }
}

__global__ __launch_bounds__(256) void gnn_node_kernel(
    const float* __restrict__ x,
    const float* __restrict__ agg,
    const float* __restrict__ local,
    const float* __restrict__ out_w,
    const float* __restrict__ out_b,
    const float* __restrict__ gamma,
    const float* __restrict__ eta,
    const float* __restrict__ xi,
    const int*   __restrict__ timestep,
    float*       __restrict__ out,
    int n_nodes)
{
    __shared__ float sW[D * D];  // 64 KB, XOR-swizzled out_w
    for (int i = threadIdx.x; i < D * D; i += 256) {
        int r = i >> 7, c = i & 127;
        sW[sw_addr(r, c)] = out_w[i];
    }
    __syncthreads();

    const int t     = timestep[0];
    const float g_t = 1.f / (1.f + __expf(-gamma[t]));
    const float e_t = 1.f / (1.f + __expf(-eta[t]));
    const float ca  = (1.f - e_t) * g_t;   // folded gate * blend scalar
    const float cx  = xi[0];

    const int lane = threadIdx.x & 31;
    const int wave = threadIdx.x >> 5;
    const int half = lane >> 4;
    const int lm   = lane & 15;

    float ob[8];
#pragma unroll
    for (int n = 0; n < 8; ++n) ob[n] = out_b[lm + (n << 4)];

    const int n_tiles = n_nodes >> 4;
    const int n_pairs = (n_tiles + 1) >> 1;
    for (int pair = blockIdx.x * 8 + wave; pair < n_pairs; pair += gridDim.x * 8) {
        const int tile0 = pair << 1;
        const int tile1 = tile0 + 1;
        const bool has1 = tile1 < n_tiles;
        const int nb0 = tile0 << 4;
        const int nb1 = (has1 ? tile1 : tile0) << 4;

        const size_t r0 = (size_t)(nb0 + lm) * D + (half << 1);
        const size_t r1 = (size_t)(nb1 + lm) * D + (half << 1);

        v8f acc0[8], acc1[8];
#pragma unroll
        for (int n = 0; n < 8; ++n) {
#pragma unroll
            for (int q = 0; q < 8; ++q) { acc0[n][q] = 0.f; acc1[n][q] = 0.f; }
        }

        v2f bb[2][8];
        v2f aa0[2], aa1[2];
#pragma unroll
        for (int n = 0; n < 8; ++n) bb[0][n] = ldsB(sW, 0, half, lm, n);
        aa0[0] = upd_frag(x, agg, local, r0, ca, cx);
        aa1[0] = upd_frag(x, agg, local, r1, ca, cx);

#pragma unroll 2
        for (int kk = 0; kk < 32; ++kk) {
            const int cur = kk & 1, nxt = cur ^ 1;
            if (kk < 31) {
#pragma unroll
                for (int n = 0; n < 8; ++n) bb[nxt][n] = ldsB(sW, kk + 1, half, lm, n);
                aa0[nxt] = upd_frag(x, agg, local, r0 + (size_t)((kk + 1) << 2), ca, cx);
                aa1[nxt] = upd_frag(x, agg, local, r1 + (size_t)((kk + 1) << 2), ca, cx);
            }
#pragma unroll
            for (int n = 0; n < 8; ++n) acc0[n] = wmma_f32(aa0[cur], bb[cur][n], acc0[n]);
#pragma unroll
            for (int n = 0; n < 8; ++n) acc1[n] = wmma_f32(aa1[cur], bb[cur][n], acc1[n]);
        }

        node_store(acc0, x, agg, local, out, nb0, half, lm, ob, ca, cx);
        if (has1) node_store(acc1, x, agg, local, out, nb1, half, lm, ob, ca, cx);
    }
}

// ---------------------------------------------------------------------------
extern "C" void kernel_launch(void* const* d_in, const int* in_sizes, int n_in,
                              void* d_out, int out_size, void* d_ws, size_t ws_size,
                              hipStream_t stream) {
    const float* x        = (const float*)d_in[0];
    const int*   src_idx  = (const int*)  d_in[1];
    const int*   dst_idx  = (const int*)  d_in[2];
    const float* local    = (const float*)d_in[3];
    const int*   timestep = (const int*)  d_in[4];
    const float* gamma    = (const float*)d_in[5];
    const float* eta      = (const float*)d_in[6];
    const float* xi       = (const float*)d_in[7];
    const float* W_w      = (const float*)d_in[8];
    const float* W_b      = (const float*)d_in[9];
    const float* ln_g     = (const float*)d_in[10];
    const float* ln_b     = (const float*)d_in[11];
    const float* out_w    = (const float*)d_in[12];
    const float* out_b    = (const float*)d_in[13];
    float*       out      = (float*)d_out;
    float*       agg      = (float*)d_ws;   // N_NODES * D floats (25.6 MB)

    const int n_nodes = in_sizes[0] / D;    // 50000
    const int n_edges = in_sizes[1];        // 600000

    zero_f4_kernel<<<512, 256, 0, stream>>>((float4*)agg, (n_nodes * D) / 4);

    // 600000/32 = 18750 tile-pairs; 1024 blocks x 8 waves -> ~2.3 pairs/wave
    gnn_edge_kernel<<<1024, 256, 0, stream>>>(
        x, src_idx, dst_idx, W_w, W_b, ln_g, ln_b, agg, n_edges);

    const int node_pairs  = ((n_nodes >> 4) + 1) >> 1;         // 1563
    const int node_blocks = (node_pairs + 7) / 8;              // 196
    gnn_node_kernel<<<node_blocks, 256, 0, stream>>>(
        x, agg, local, out_w, out_b, gamma, eta, xi, timestep, out, n_nodes);
}